// ManifoldConstrainedHyperConnections_13606456394129
// MI455X (gfx1250) — compile-verified
//
#include <hip/hip_runtime.h>
#include <math.h>
#include <stdint.h>

// ---- problem constants (S=4, V=1, FR=1, DIM=2048, B=4, N=2048) ----
#define SS    4
#define TT    5
#define DIMD  2048
#define KK    (SS * DIMD)      // 8192 : reduced dim of the per-site GEMV
#define NT    8                // sites per workgroup  (8*8192*4B = 256KB LDS tile)
#define NCOL  24               // 20 alpha cols + 4 beta cols
#define NWAVES 8               // 256 threads / wave32
#define KPW   (KK / NWAVES)    // 1024 K per wave

typedef __attribute__((ext_vector_type(2))) float v2f;
typedef __attribute__((ext_vector_type(4))) float v4f;
typedef __attribute__((ext_vector_type(8))) float v8f;

struct __align__(16) Smem {
  float xs[NT * KK];     // 256 KB  raw residual tile, xs[m][k], k = s*2048+dd
  float g1[KK];          //  32 KB  (gamma + 1)
  float dots[NT * 32];   //   1 KB  accumulated dot products (24 of 32 used)
  float coef[NT * 16];   //  512 B  final 4x4 mixing coefs per site
  float ss[NT];          //  sum of squares per site
};

// ---------------------------------------------------------------------------
// Prep: pack the 8192x20 alpha + 8192x4 beta weights (+8 zero pad cols) into
// the exact WMMA-B operand layout:  Wpk[tile][k>>1][col] = {W[k][c], W[k+1][c]}
// with c = tile*16+col.  1 MB total -> L2 resident, loads become one
// coalesced global_load_b64 per tile per WMMA step, branch-free.
// ---------------------------------------------------------------------------
__global__ __launch_bounds__(256)
void pack_weights(const float* __restrict__ Wa, const float* __restrict__ Wb,
                  float* __restrict__ wpk) {
  int id  = blockIdx.x * 256 + threadIdx.x;   // 131072 threads: (k, col)
  int k   = id >> 4;
  int col = id & 15;
  #pragma unroll
  for (int t = 0; t < 2; ++t) {
    int c = t * 16 + col;
    float v = (c < 20) ? Wa[k * 20 + c]
            : (c < 24) ? Wb[k * 4 + (c - 20)]
            : 0.0f;
    wpk[((t * (KK/2) + (k >> 1)) * 16 + col) * 2 + (k & 1)] = v;
  }
}

__global__ __launch_bounds__(256)
void mchc_kernel(const float* __restrict__ resid,   // (16, 2048, 2048)
                 const float* __restrict__ gamma,   // (8192,)
                 const float* __restrict__ wpk,     // packed weights (1MB)
                 const float* __restrict__ sa,      // (4, 5)
                 const float* __restrict__ pbs,     // (1,)
                 const float* __restrict__ rs,      // (1,)
                 const float* __restrict__ sb,      // (4,)
                 const float* __restrict__ hps,     // (1,)
                 float* __restrict__ out)           // (16, 2048, 2048)
{
  __shared__ Smem sm;
  const int tid = threadIdx.x;
  const int wg  = blockIdx.x;
  const int b   = wg >> 8;             // 4 batches
  const int n0  = (wg & 255) * NT;     // 256 n-tiles of 8 sites

  // ---- zero accumulators, stage (gamma+1) into LDS ----
  if (tid < NT * 32) sm.dots[tid] = 0.0f;
  if (tid < NT)      sm.ss[tid]   = 0.0f;
  {
    const v4f* g4 = (const v4f*)gamma;
    v4f*       o4 = (v4f*)sm.g1;
    #pragma unroll
    for (int i = 0; i < (KK / 4) / 256; ++i) {     // 8 iters
      int q = i * 256 + tid;
      o4[q] = g4[q] + 1.0f;
    }
  }

  // ---- Phase 1: async global->LDS copy of the residual tile (ASYNCcnt path).
  //      Non-temporal: this stream is touched once, keep L2 for the weights. ----
  for (int m = 0; m < NT; ++m) {
    const int n = n0 + m;
    #pragma unroll
    for (int i = 0; i < 8; ++i) {                  // 2048 float4 / 256 threads
      int q   = i * 256 + tid;                     // [0,2048)
      int s   = q >> 9;                            // shard
      int dd4 = q & 511;
      const float* gsrc = resid + (((size_t)(b * SS + s) * 2048 + n) * 2048) + dd4 * 4;
      // flat shared addr truncated to 32 bits == LDS byte offset (ISA aperture rule)
      unsigned lds = (unsigned)(uintptr_t)(sm.xs + m * KK + q * 4);
      unsigned long long ga = (unsigned long long)(uintptr_t)gsrc;
      asm volatile("global_load_async_to_lds_b128 %0, %1, off th:TH_LOAD_NT"
                   :: "v"(lds), "v"(ga) : "memory");
    }
  }
  asm volatile("s_wait_asynccnt 0x0" ::: "memory");
  __syncthreads();

  // sum of squares from the LDS tile
  for (int m = 0; m < NT; ++m) {
    float acc = 0.0f;
    const v4f* xs4 = (const v4f*)(sm.xs + m * KK);
    #pragma unroll
    for (int i = 0; i < 8; ++i) {
      v4f x = xs4[i * 256 + tid];
      acc += x.x * x.x + x.y * x.y + x.z * x.z + x.w * x.w;
    }
    atomicAdd(&sm.ss[m], acc);
  }
  __syncthreads();

  // ---- Phase 2: fp32 WMMA GEMV  dots[site][col] = sum_k xs[site][k]*(1+g[k])*W[k][col] ----
  // M=16: rows 8..15 are harmless duplicates of rows 0..7 (read row mrow&7,
  // finite data, rows of D are independent; we only consume rows 0..7).
  const int wave = tid >> 5;
  const int lane = tid & 31;
  const int mrow = lane & 15;
  const int kh   = lane >> 4;          // upper half-lanes take K+2,K+3
  const int col  = lane & 15;

  v8f c0 = {}; v8f c1 = {};
  const int kbase = wave * KPW;
  const v2f* xsrow = (const v2f*)(sm.xs + (mrow & 7) * KK);
  const v2f* g2    = (const v2f*)sm.g1;
  const v2f* wp    = (const v2f*)wpk;
  const int idx2base = (kbase >> 1) + kh;

  #pragma unroll 4
  for (int i = 0; i < KPW / 4; ++i) {              // 256 WMMA steps per wave
    int idx2 = idx2base + 2 * i;                   // (k0 + 2*kh) >> 1
    v2f xv = xsrow[idx2];
    v2f gv = g2[idx2];
    v2f a = xv * gv;

    v2f b0 = wp[idx2 * 16 + col];                  // tile0 (alpha cols 0..15)
    v2f b1 = wp[(KK/2 + idx2) * 16 + col];         // tile1 (cols 16..23 + pad)

    c0 = __builtin_amdgcn_wmma_f32_16x16x4_f32(false, a, false, b0, (short)0, c0, false, false);
    c1 = __builtin_amdgcn_wmma_f32_16x16x4_f32(false, a, false, b1, (short)0, c1, false, false);
  }

  // cross-wave reduction of partial C tiles (rows 0..7 live in lanes 0..15, VGPRs 0..7)
  if (lane < 16) {
    #pragma unroll
    for (int j = 0; j < NT; ++j) {
      atomicAdd(&sm.dots[j * 32 + col],      c0[j]);
      atomicAdd(&sm.dots[j * 32 + 16 + col], c1[j]);
    }
  }
  __syncthreads();

  // ---- Phase 3: per-site scalar tail — norm scale, sigmoid, 4x4 Sinkhorn, coefs ----
  if (tid < NT) {
    const int msite = tid;
    float rn = sqrtf((float)KK) / fmaxf(sqrtf(sm.ss[msite]), 1e-12f);
    float dv[NCOL];
    #pragma unroll
    for (int c = 0; c < NCOL; ++c) dv[c] = sm.dots[msite * 32 + c] * rn;

    const float pbs0 = pbs[0], rs0 = rs[0], hps0 = hps[0];

    float apre[4], M[4][4];
    #pragma unroll
    for (int s = 0; s < 4; ++s) {
      float a0 = dv[s * 5 + 0] * pbs0 + sa[s * 5 + 0];
      apre[s] = 1.0f / (1.0f + expf(-a0));
      #pragma unroll
      for (int t = 0; t < 4; ++t)
        M[s][t] = dv[s * 5 + 1 + t] * rs0 + sa[s * 5 + 1 + t];
    }
    for (int it = 0; it < 20; ++it) {
      #pragma unroll
      for (int t = 0; t < 4; ++t) {          // logsumexp over s (columns)
        float mx = fmaxf(fmaxf(M[0][t], M[1][t]), fmaxf(M[2][t], M[3][t]));
        float l  = mx + logf(expf(M[0][t]-mx) + expf(M[1][t]-mx) +
                             expf(M[2][t]-mx) + expf(M[3][t]-mx));
        #pragma unroll
        for (int s = 0; s < 4; ++s) M[s][t] -= l;
      }
      #pragma unroll
      for (int s = 0; s < 4; ++s) {          // logsumexp over t (rows)
        float mx = fmaxf(fmaxf(M[s][0], M[s][1]), fmaxf(M[s][2], M[s][3]));
        float l  = mx + logf(expf(M[s][0]-mx) + expf(M[s][1]-mx) +
                             expf(M[s][2]-mx) + expf(M[s][3]-mx));
        #pragma unroll
        for (int t = 0; t < 4; ++t) M[s][t] -= l;
      }
    }
    float beta[4];
    #pragma unroll
    for (int s = 0; s < 4; ++s) {
      float x = dv[20 + s] * hps0 + sb[s];
      beta[s] = 2.0f / (1.0f + expf(-x));
    }
    // out[os] = beta[os]*mix0 + mix_{1+os} = sum_s' coef[os][s'] * r[s']
    #pragma unroll
    for (int os = 0; os < 4; ++os)
      #pragma unroll
      for (int sp = 0; sp < 4; ++sp)
        sm.coef[msite * 16 + os * 4 + sp] = beta[os] * apre[sp] + expf(M[sp][os]);
  }
  __syncthreads();

  // ---- Phase 4: mix from LDS, stream output with non-temporal b128 stores ----
  for (int mm = 0; mm < NT; ++mm) {
    const int n = n0 + mm;
    const v4f* xrow = (const v4f*)(sm.xs + mm * KK);
    float cf[16];
    #pragma unroll
    for (int i = 0; i < 16; ++i) cf[i] = sm.coef[mm * 16 + i];
    #pragma unroll
    for (int j = 0; j < 2; ++j) {
      int q = j * 256 + tid;                 // float4 index in [0,512) over d
      v4f x0 = xrow[0 * 512 + q];
      v4f x1 = xrow[1 * 512 + q];
      v4f x2 = xrow[2 * 512 + q];
      v4f x3 = xrow[3 * 512 + q];
      #pragma unroll
      for (int os = 0; os < 4; ++os) {
        v4f o = cf[os*4+0] * x0 + cf[os*4+1] * x1 + cf[os*4+2] * x2 + cf[os*4+3] * x3;
        v4f* dst = (v4f*)(out + (((size_t)(b * SS + os) * 2048 + n) * 2048));
        __builtin_nontemporal_store(o, dst + q);
      }
    }
  }
}

extern "C" void kernel_launch(void* const* d_in, const int* in_sizes, int n_in,
                              void* d_out, int out_size, void* d_ws, size_t ws_size,
                              hipStream_t stream) {
  const float* resid = (const float*)d_in[0];
  const float* gamma = (const float*)d_in[1];
  const float* Wa    = (const float*)d_in[2];
  const float* sa    = (const float*)d_in[3];
  const float* pbs   = (const float*)d_in[4];
  const float* rs    = (const float*)d_in[5];
  const float* Wb    = (const float*)d_in[6];
  const float* sb    = (const float*)d_in[7];
  const float* hps   = (const float*)d_in[8];
  float* outp = (float*)d_out;
  float* wpk  = (float*)d_ws;   // needs 2*8192*16*4 = 1 MB of scratch

  // pack weights into WMMA-B layout (runs every call: deterministic)
  hipLaunchKernelGGL(pack_weights, dim3((KK * 16) / 256), dim3(256), 0, stream,
                     Wa, Wb, wpk);

  dim3 grid(4 * 256);   // B * (N / NT)
  dim3 block(256);      // 8 wave32
  hipLaunchKernelGGL(mchc_kernel, grid, block, 0, stream,
                     resid, gamma, wpk, sa, pbs, rs, sb, hps, outp);
}